// LWR_69166153335081
// MI455X (gfx1250) — compile-verified
//
#include <hip/hip_runtime.h>
#include <hip/hip_bf16.h>
#include <math.h>

// ---------------- constants from the reference ----------------
#define BATCH       1024
#define D_IN        256
#define DK          128
#define NCLS        100
#define TEA         4
#define DSLEN       200000
#define KEPOCH      30
#define MAXEP       150
#define TAU         3.0f
#define UPD         0.9f

typedef __attribute__((ext_vector_type(2))) float v2f;
typedef __attribute__((ext_vector_type(8))) float v8f;

// =====================================================================
// Kernel 1: q[1024,128] = query[1024,256] @ Wq[128,256]^T + bq
// One wave computes a 16x32 output tile (two 16x16 WMMA accumulators
// sharing each A fragment -> 2x arithmetic intensity on A, and two
// independent V_WMMA_F32_16X16X4_F32 dependency chains).
// A (16x4 f32) per-lane layout: lanes 0-15 hold (K=0,K=1) for M=lane,
// lanes 16-31 hold (K=2,K=3) for M=lane-16.  B (4x16) mirrors that with
// N striped across lanes.  C/D: VGPR i, lanes 0-15 -> M=i, 16-31 -> M=i+8.
// =====================================================================
__global__ void gemm_q_kernel(const float* __restrict__ query,
                              const float* __restrict__ Wq,
                              const float* __restrict__ bq,
                              float* __restrict__ q) {
    const int lane = threadIdx.x & 31;
    const int wave = threadIdx.x >> 5;
    const int p    = blockIdx.x * 4 + wave;          // 256 tile-pairs = 64 Mtiles * 4 Npairs
    const int tM   = (p >> 2) << 4;
    const int tN   = (p & 3) << 5;                   // 32-wide N strip
    const int half = lane >> 4;                      // 0: K=0,1   1: K=2,3
    const int l    = lane & 15;

    const float* __restrict__ arow  = query + (size_t)(tM + l) * D_IN + half * 2;
    const float* __restrict__ brow0 = Wq + (size_t)(tN + l)      * D_IN + half * 2; // B[k][n]=Wq[n][k]
    const float* __restrict__ brow1 = Wq + (size_t)(tN + 16 + l) * D_IN + half * 2;

    v8f acc0 = {};
    v8f acc1 = {};
    for (int k = 0; k < D_IN; k += 4) {
        v2f a, b0, b1;
        a.x  = arow[k];   a.y  = arow[k + 1];
        b0.x = brow0[k];  b0.y = brow0[k + 1];
        b1.x = brow1[k];  b1.y = brow1[k + 1];
        acc0 = __builtin_amdgcn_wmma_f32_16x16x4_f32(
                   false, a, false, b0, (short)0, acc0, false, false);
        acc1 = __builtin_amdgcn_wmma_f32_16x16x4_f32(
                   false, a, false, b1, (short)0, acc1, false, false);
    }

    const float bias0 = bq[tN + l];
    const float bias1 = bq[tN + 16 + l];
#pragma unroll
    for (int i = 0; i < 8; ++i) {
        const int row = tM + i + half * 8;
        q[(size_t)row * DK + tN + l]      = acc0[i] + bias0;
        q[(size_t)row * DK + tN + 16 + l] = acc1[i] + bias1;
    }
}

// =====================================================================
// Kernel 2: r[1024,256] = q[1024,128] @ Wk[128,256]   (Wk row-major [k][d])
// Same 16x32-per-wave scheme.
// =====================================================================
__global__ void gemm_r_kernel(const float* __restrict__ q,
                              const float* __restrict__ Wk,
                              float* __restrict__ r) {
    const int lane = threadIdx.x & 31;
    const int wave = threadIdx.x >> 5;
    const int p    = blockIdx.x * 4 + wave;          // 512 tile-pairs = 64 Mtiles * 8 Npairs
    const int tM   = (p >> 3) << 4;
    const int tN   = (p & 7) << 5;
    const int half = lane >> 4;
    const int l    = lane & 15;

    const float* __restrict__ arow = q + (size_t)(tM + l) * DK + half * 2;

    v8f acc0 = {};
    v8f acc1 = {};
    for (int k = 0; k < DK; k += 4) {
        const int kk = k + half * 2;
        v2f a, b0, b1;
        a.x  = arow[k];  a.y = arow[k + 1];
        b0.x = Wk[(size_t)kk * D_IN + tN + l];
        b0.y = Wk[(size_t)(kk + 1) * D_IN + tN + l];
        b1.x = Wk[(size_t)kk * D_IN + tN + 16 + l];
        b1.y = Wk[(size_t)(kk + 1) * D_IN + tN + 16 + l];
        acc0 = __builtin_amdgcn_wmma_f32_16x16x4_f32(
                   false, a, false, b0, (short)0, acc0, false, false);
        acc1 = __builtin_amdgcn_wmma_f32_16x16x4_f32(
                   false, a, false, b1, (short)0, acc1, false, false);
    }

#pragma unroll
    for (int i = 0; i < 8; ++i) {
        const int row = tM + i + half * 8;
        r[(size_t)row * D_IN + tN + l]      = acc0[i];
        r[(size_t)row * D_IN + tN + 16 + l] = acc1[i];
    }
}

// =====================================================================
// Block reductions (blockDim.x == 128)
// =====================================================================
__device__ __forceinline__ float blk_sum128(float* s, int tid, float v) {
    s[tid] = v; __syncthreads();
    for (int off = 64; off > 0; off >>= 1) {
        if (tid < off) s[tid] += s[tid + off];
        __syncthreads();
    }
    float res = s[0]; __syncthreads();
    return res;
}
__device__ __forceinline__ float blk_max128(float* s, int tid, float v) {
    s[tid] = v; __syncthreads();
    for (int off = 64; off > 0; off >>= 1) {
        if (tid < off) s[tid] = fmaxf(s[tid], s[tid + off]);
        __syncthreads();
    }
    float res = s[0]; __syncthreads();
    return res;
}

// =====================================================================
// Kernel 3: per batch-row fused epilogue.
//   energy[t]  = r[b,:] . keys_bank[t, idx, :]      (q.bk cancels in softmax)
//   attn       = softmax_t(energy)
//   ft[c]      = sum_t attn[t] * values_bank[t, idx, c]
//   p_t        = softmax(ft / TAU)         -> d_out
//   loss1_part = logsumexp(logits[b]) - logits[b, y]
//   loss2_part = sum_c p_t*(log_pt - log_softmax(logits[b]/TAU))
// grid = 1024 blocks, block = 128 threads
// =====================================================================
__global__ void epilogue_kernel(const int*   __restrict__ batch_idx,
                                const float* __restrict__ logits,
                                const int*   __restrict__ y_true,
                                const int*   __restrict__ cur_epoch_p,
                                const float* __restrict__ keys_bank,
                                const float* __restrict__ values_bank,
                                const float* __restrict__ r,
                                float* __restrict__ p_t_out,
                                float* __restrict__ loss1_part,
                                float* __restrict__ loss2_part) {
    const int b   = blockIdx.x;
    const int tid = threadIdx.x;

    __shared__ float s_red[128];
    __shared__ float s_attn[TEA];
    __shared__ float s_ft[NCLS];     // final_teacher / TAU

    int ep = *cur_epoch_p;
    int cur_tea = (ep - 1) / KEPOCH;
    if (cur_tea > TEA) cur_tea = TEA;
    if (cur_tea < 1)   cur_tea = 1;

    const int idx = batch_idx[b];

    // ---- teacher energies: 4 dot products of length 256 ----
    float e[TEA] = {0.f, 0.f, 0.f, 0.f};
    for (int d = tid; d < D_IN; d += 128) {
        const float rv = r[(size_t)b * D_IN + d];
        for (int t = 0; t < cur_tea; ++t)
            e[t] += rv * keys_bank[((size_t)t * DSLEN + idx) * D_IN + d];
    }
    for (int t = 0; t < cur_tea; ++t) {
        const float et = blk_sum128(s_red, tid, e[t]);
        if (tid == 0) s_attn[t] = et;
    }
    __syncthreads();

    // ---- softmax over teachers (serial, 4 elements) ----
    if (tid == 0) {
        float m = s_attn[0];
        for (int t = 1; t < cur_tea; ++t) m = fmaxf(m, s_attn[t]);
        float sum = 0.f;
        for (int t = 0; t < cur_tea; ++t) {
            const float v = __expf(s_attn[t] - m);
            s_attn[t] = v; sum += v;
        }
        const float inv = 1.0f / sum;
        for (int t = 0; t < cur_tea; ++t) s_attn[t] *= inv;
    }
    __syncthreads();

    // ---- final teacher logits / TAU ----
    if (tid < NCLS) {
        float ft = 0.f;
        for (int t = 0; t < cur_tea; ++t)
            ft += s_attn[t] * values_bank[((size_t)t * DSLEN + idx) * NCLS + tid];
        s_ft[tid] = ft * (1.0f / TAU);
    }
    __syncthreads();

    const float lg = (tid < NCLS) ? logits[(size_t)b * NCLS + tid] : 0.f;
    const float lq = lg * (1.0f / TAU);

    // ---- three logsumexps over 100 classes ----
    const float m_ft = blk_max128(s_red, tid, (tid < NCLS) ? s_ft[tid] : -1e30f);
    const float S_ft = blk_sum128(s_red, tid, (tid < NCLS) ? __expf(s_ft[tid] - m_ft) : 0.f);
    const float m_lq = blk_max128(s_red, tid, (tid < NCLS) ? lq : -1e30f);
    const float S_lq = blk_sum128(s_red, tid, (tid < NCLS) ? __expf(lq - m_lq) : 0.f);
    const float m_l  = blk_max128(s_red, tid, (tid < NCLS) ? lg : -1e30f);
    const float S_l  = blk_sum128(s_red, tid, (tid < NCLS) ? __expf(lg - m_l) : 0.f);

    float part2 = 0.f;
    if (tid < NCLS) {
        const float pt     = __expf(s_ft[tid] - m_ft) / S_ft;
        const float log_pt = (s_ft[tid] - m_ft) - __logf(S_ft);
        const float log_q  = lq - (m_lq + __logf(S_lq));
        part2 = pt * (log_pt - log_q);
        p_t_out[(size_t)b * NCLS + tid] = pt;
    }
    const float sum2 = blk_sum128(s_red, tid, part2);

    if (tid == 0) {
        loss2_part[b] = sum2;
        const float lse = m_l + __logf(S_l);
        loss1_part[b] = lse - logits[(size_t)b * NCLS + y_true[b]];
    }
}

// =====================================================================
// Kernel 4: reduce the 1024 per-row partials, apply alpha scaling.
// =====================================================================
__global__ void finalize_kernel(const float* __restrict__ loss1_part,
                                const float* __restrict__ loss2_part,
                                const int*   __restrict__ cur_epoch_p,
                                float* __restrict__ out) {
    __shared__ float s1[256];
    __shared__ float s2[256];
    const int tid = threadIdx.x;
    float a1 = 0.f, a2 = 0.f;
    for (int i = tid; i < BATCH; i += 256) { a1 += loss1_part[i]; a2 += loss2_part[i]; }
    s1[tid] = a1; s2[tid] = a2; __syncthreads();
    for (int off = 128; off > 0; off >>= 1) {
        if (tid < off) { s1[tid] += s1[tid + off]; s2[tid] += s2[tid + off]; }
        __syncthreads();
    }
    if (tid == 0) {
        const int ep = *cur_epoch_p;
        const float alpha = 1.0f - UPD * (float)(ep - ep % KEPOCH) / (float)MAXEP;
        out[0] = alpha * (s1[0] / (float)BATCH);
        out[1] = (1.0f - alpha) * (TAU * TAU) * (s2[0] / (float)BATCH);
    }
}

// =====================================================================
// launch
// =====================================================================
extern "C" void kernel_launch(void* const* d_in, const int* in_sizes, int n_in,
                              void* d_out, int out_size, void* d_ws, size_t ws_size,
                              hipStream_t stream) {
    const int*   batch_idx   = (const int*)  d_in[0];
    const float* query       = (const float*)d_in[1];
    const float* logits      = (const float*)d_in[2];
    const int*   y_true      = (const int*)  d_in[3];
    const int*   cur_epoch_p = (const int*)  d_in[4];
    const float* keys_bank   = (const float*)d_in[5];
    const float* values_bank = (const float*)d_in[6];
    const float* Wq          = (const float*)d_in[7];
    const float* bq          = (const float*)d_in[8];
    const float* Wk          = (const float*)d_in[9];
    // bk (d_in[10]) cancels inside the teacher softmax -> unused.

    float* out    = (float*)d_out;          // [0]=loss1, [1]=loss2, [2..]=p_t
    float* ws     = (float*)d_ws;
    float* q_buf  = ws;                                   // 1024*128
    float* r_buf  = ws + (size_t)BATCH * DK;              // 1024*256
    float* l1p    = r_buf + (size_t)BATCH * D_IN;         // 1024
    float* l2p    = l1p + BATCH;                          // 1024

    // 1) q = query @ Wq^T + bq : 256 16x32 tiles, 4 waves/block
    gemm_q_kernel<<<64, 128, 0, stream>>>(query, Wq, bq, q_buf);

    // 2) r = q @ Wk : 512 16x32 tiles, 4 waves/block
    gemm_r_kernel<<<128, 128, 0, stream>>>(q_buf, Wk, r_buf);

    // 3) fused attention/softmax/KL epilogue, one block per batch row
    epilogue_kernel<<<BATCH, 128, 0, stream>>>(batch_idx, logits, y_true, cur_epoch_p,
                                               keys_bank, values_bank, r_buf,
                                               out + 2, l1p, l2p);

    // 4) scalar losses
    finalize_kernel<<<1, 256, 0, stream>>>(l1p, l2p, cur_epoch_p, out);
}